// GNN_70858370449990
// MI455X (gfx1250) — compile-verified
//
#include <hip/hip_runtime.h>

// ---------------- problem constants ----------------
#define BATCH   64
#define NTOK    577
#define PTOK    576       // NTOK-1
#define DIM     768
#define DENS    288       // PTOK * KEEP_RATIO
#define SSZ     288       // PTOK - DENS
#define GSEL    144       // SSZ * GROUP_RATIO
#define OUTROWS 433       // 1 + DENS + GSEL
#define RCOLS   432       // DENS + GSEL
#define MWORDS  9         // 288/32 mask words per row

typedef float v2f __attribute__((ext_vector_type(2)));
typedef float v8f __attribute__((ext_vector_type(8)));

// ---------------------------------------------------------------------------
// K1: per-batch descending argsort of cls attention (rank-by-count, stable)
// order[rank] = i  with rank = #{j: a[j]>a[i]} + #{j: a[j]==a[i], j<i}
// ---------------------------------------------------------------------------
__global__ __launch_bounds__(PTOK)
void order_kernel(const float* __restrict__ attn, int* __restrict__ order) {
    const int b = blockIdx.x, t = threadIdx.x;
    __shared__ float a[PTOK];
    a[t] = attn[(size_t)b * NTOK * PTOK + t];   // attention[b, 0, t]
    __syncthreads();
    const float v = a[t];
    int rank = 0;
    for (int j = 0; j < PTOK; ++j) {
        const float w = a[j];
        rank += (w > v) || (w == v && j < t);
    }
    order[b * PTOK + rank] = t;
}

// ---------------------------------------------------------------------------
// K2: adjacency of the skip half, diag zeroed.
// adj[b,i,j] = attention[b, 1+order[288+i], order[288+j]] ; adj[i,i]=0
// ---------------------------------------------------------------------------
__global__ __launch_bounds__(SSZ)
void adj_kernel(const float* __restrict__ attn, const int* __restrict__ order,
                float* __restrict__ W) {
    const int b = blockIdx.y, i = blockIdx.x, j = threadIdx.x;
    __shared__ int ocol[SSZ];
    ocol[j] = order[b * PTOK + DENS + j];
    __syncthreads();
    const int srow = 1 + order[b * PTOK + DENS + i];
    float v = 0.0f;
    if (i != j)
        v = attn[(size_t)b * NTOK * PTOK + (size_t)srow * PTOK + ocol[j]];
    W[((size_t)b * SSZ + i) * SSZ + j] = v;
}

// ---------------------------------------------------------------------------
// K3: gather tokens. r==0 -> cls to out; 1..288 -> non_skip to out;
// 289..576 -> skip features to scratch. 192 threads x float4 = 768 floats.
// ---------------------------------------------------------------------------
__global__ __launch_bounds__(192)
void gather_tokens_kernel(const float* __restrict__ tk, const int* __restrict__ order,
                          float* __restrict__ out_tokens, float* __restrict__ skip) {
    const int b = blockIdx.y, r = blockIdx.x, t = threadIdx.x;
    const int src = (r == 0) ? 0 : (1 + order[b * PTOK + (r - 1)]);
    const float4* s = (const float4*)(tk + ((size_t)b * NTOK + src) * DIM);
    float4* d;
    if (r <= DENS) d = (float4*)(out_tokens + ((size_t)b * OUTROWS + r) * DIM);
    else           d = (float4*)(skip + ((size_t)b * SSZ + (r - DENS - 1)) * DIM);
    d[t] = s[t];
}

// ---------------------------------------------------------------------------
// K4: top-2 per row (stable: strict >, earlier index wins ties)
// ---------------------------------------------------------------------------
__global__ __launch_bounds__(SSZ)
void top2_kernel(const float* __restrict__ W, int* __restrict__ t0, int* __restrict__ t1) {
    const int b = blockIdx.x, i = threadIdx.x;
    const float* row = W + ((size_t)b * SSZ + i) * SSZ;
    float v0 = -1.0f, v1 = -1.0f;
    int i0 = 0, i1 = 0;
    for (int j = 0; j < SSZ; ++j) {
        const float v = row[j];
        if (v > v0)      { v1 = v0; i1 = i0; v0 = v; i0 = j; }
        else if (v > v1) { v1 = v;  i1 = j; }
    }
    t0[b * SSZ + i] = i0;
    t1[b * SSZ + i] = i1;
}

// ---------------------------------------------------------------------------
// K5: symmetrized top-k binary mask applied to W in place
// ---------------------------------------------------------------------------
__global__ __launch_bounds__(SSZ)
void mask_topk_kernel(float* __restrict__ W, const int* __restrict__ t0,
                      const int* __restrict__ t1) {
    const int b = blockIdx.y, i = blockIdx.x, j = threadIdx.x;
    const int a0 = t0[b * SSZ + i], a1 = t1[b * SSZ + i];
    const int b0 = t0[b * SSZ + j], b1 = t1[b * SSZ + j];
    const bool keep = (j == a0) || (j == a1) || (i == b0) || (i == b1);
    if (!keep) W[((size_t)b * SSZ + i) * SSZ + j] = 0.0f;
}

// ---------------------------------------------------------------------------
// K6: device-side fixpoint (KthAverage). One block per batch, mask as LDS
// bitset (288x288 bits = 10.4 KB). Mask is monotone-shrinking, so per-batch
// convergence == reference's global while_loop fixpoint. Deterministic.
// Afterwards: deg[j] = final column sums, and W pruned in place.
// ---------------------------------------------------------------------------
__global__ __launch_bounds__(SSZ)
void fixpoint_kernel(float* __restrict__ W, float* __restrict__ deg_out) {
    const int b = blockIdx.x, t = threadIdx.x;
    __shared__ unsigned mask[SSZ * MWORDS];
    __shared__ float avg[SSZ];
    __shared__ int changed;
    float* Wb = W + (size_t)b * SSZ * SSZ;

    // init: thread t owns row t of the mask
    for (int w = 0; w < MWORDS; ++w) {
        unsigned bits = 0;
        const float* rp = Wb + (size_t)t * SSZ + w * 32;
        for (int j = 0; j < 32; ++j)
            if (rp[j] != 0.0f) bits |= (1u << j);
        mask[t * MWORDS + w] = bits;
    }
    __syncthreads();

    int lastDeg = 0;
    const int wsel = t >> 5;
    const unsigned bsel = 1u << (t & 31);
    for (;;) {
        // column t: degree + weighted sum -> avg
        float s = 0.0f; int d = 0;
        for (int i = 0; i < SSZ; ++i) {
            if (mask[i * MWORDS + wsel] & bsel) { s += Wb[(size_t)i * SSZ + t]; ++d; }
        }
        avg[t] = (d > 0) ? s / (float)d : 0.0f;
        lastDeg = d;
        if (t == 0) changed = 0;
        __syncthreads();                    // avg + reset visible

        // row t: m[t][j] &= (avg[j] > avg[t])
        const float ai = avg[t];
        int ch = 0;
        for (int w = 0; w < MWORDS; ++w) {
            const unsigned bits = mask[t * MWORDS + w];
            unsigned nb = bits, rem = bits;
            while (rem) {
                const int j = __ffs(rem) - 1;
                rem &= rem - 1;
                if (!(avg[w * 32 + j] > ai)) nb &= ~(1u << j);
            }
            if (nb != bits) { mask[t * MWORDS + w] = nb; ch = 1; }
        }
        if (ch) atomicAdd(&changed, 1);
        __syncthreads();                    // updates + changed visible
        const int c = changed;
        __syncthreads();                    // all reads done before next reset
        if (c == 0) break;
    }

    deg_out[b * SSZ + t] = (float)lastDeg;
    // prune W row t by final mask
    for (int w = 0; w < MWORDS; ++w) {
        const unsigned bits = mask[t * MWORDS + w];
        for (int j = 0; j < 32; ++j)
            if (!(bits & (1u << j))) Wb[(size_t)t * SSZ + w * 32 + j] = 0.0f;
    }
}

// ---------------------------------------------------------------------------
// K7: WMMA aggregation: Y[j,d] = sum_i Wf[i,j] * X[i,d] + X[j,d]
// One wave per 16x16 output tile; K-loop over i in steps of 4 with
// V_WMMA_F32_16X16X4_F32 (full f32 precision, inputs are f32).
// A layout (ISA 7.12.2, 32-bit A 16x4): lane = half*16 + m;
//   VGPR0 = A[m][2*half], VGPR1 = A[m][2*half+1]
// C/D layout: acc[v] = D[v + 8*half][lane%16]
// ---------------------------------------------------------------------------
__global__ __launch_bounds__(32)
void wmma_aggregate_kernel(const float* __restrict__ W, const float* __restrict__ X,
                           float* __restrict__ Y) {
    const int b = blockIdx.z;
    const int jBase = blockIdx.y * 16;      // M: output row j (0..287)
    const int dBase = blockIdx.x * 16;      // N: feature d   (0..767)
    const float* Wb = W + (size_t)b * SSZ * SSZ;
    const float* Xb = X + (size_t)b * SSZ * DIM;
    float* Yb = Y + (size_t)b * SSZ * DIM;
    const int lane = threadIdx.x;
    const int half = lane >> 4;
    const int mn = lane & 15;

    v8f acc = {};
    for (int k0 = 0; k0 < SSZ; k0 += 4) {
        const int ka = k0 + 2 * half;
        v2f a, bm;
        // A[m][k] = Wf[k][j]  (transposed use of W)
        a.x = Wb[(size_t)ka * SSZ + jBase + mn];
        a.y = Wb[(size_t)(ka + 1) * SSZ + jBase + mn];
        // B[k][n] = X[k][d]
        bm.x = Xb[(size_t)ka * DIM + dBase + mn];
        bm.y = Xb[(size_t)(ka + 1) * DIM + dBase + mn];
        acc = __builtin_amdgcn_wmma_f32_16x16x4_f32(
            false, a, false, bm, (short)0, acc, false, false);
    }
#pragma unroll
    for (int v = 0; v < 8; ++v) {
        const int row = jBase + v + 8 * half;
        const int col = dBase + mn;
        Yb[(size_t)row * DIM + col] = acc[v] + Xb[(size_t)row * DIM + col];
    }
}

// ---------------------------------------------------------------------------
// K8: top-144 of degree by stable rank -> group_idx
// ---------------------------------------------------------------------------
__global__ __launch_bounds__(SSZ)
void group_kernel(const float* __restrict__ deg, int* __restrict__ gidx) {
    const int b = blockIdx.x, t = threadIdx.x;
    __shared__ float d[SSZ];
    d[t] = deg[b * SSZ + t];
    __syncthreads();
    const float v = d[t];
    int rank = 0;
    for (int j = 0; j < SSZ; ++j) {
        const float w = d[j];
        rank += (w > v) || (w == v && j < t);
    }
    if (rank < GSEL) gidx[b * GSEL + rank] = t;
}

// ---------------------------------------------------------------------------
// K9: gather selected aggregated rows into out_tokens rows 289..432
// ---------------------------------------------------------------------------
__global__ __launch_bounds__(192)
void gather_sel_kernel(const float* __restrict__ sembs, const int* __restrict__ gidx,
                       float* __restrict__ out_tokens) {
    const int b = blockIdx.y, g = blockIdx.x, t = threadIdx.x;
    const int src = gidx[b * GSEL + g];
    const float4* s = (const float4*)(sembs + ((size_t)b * SSZ + src) * DIM);
    float4* d = (float4*)(out_tokens + ((size_t)b * OUTROWS + 1 + DENS + g) * DIM);
    d[t] = s[t];
}

// ---------------------------------------------------------------------------
// K10: build R[b,r,c] = attention[b, srcrow, srccol] via order/group_idx
// ---------------------------------------------------------------------------
__global__ __launch_bounds__(RCOLS)
void build_R_kernel(const float* __restrict__ attn, const int* __restrict__ order,
                    const int* __restrict__ gidx, float* __restrict__ R) {
    const int b = blockIdx.y, r = blockIdx.x, c = threadIdx.x;
    int srow;
    if (r == 0)           srow = 0;
    else if (r <= DENS)   srow = 1 + order[b * PTOK + r - 1];
    else                  srow = 1 + order[b * PTOK + DENS + gidx[b * GSEL + (r - DENS - 1)]];
    const int scol = (c < DENS) ? order[b * PTOK + c]
                                : order[b * PTOK + DENS + gidx[b * GSEL + (c - DENS)]];
    R[((size_t)b * OUTROWS + r) * RCOLS + c] =
        attn[(size_t)b * NTOK * PTOK + (size_t)srow * PTOK + scol];
}

// ---------------------------------------------------------------------------
extern "C" void kernel_launch(void* const* d_in, const int* in_sizes, int n_in,
                              void* d_out, int out_size, void* d_ws, size_t ws_size,
                              hipStream_t stream) {
    const float* tk   = (const float*)d_in[0];   // [64,577,768]
    const float* attn = (const float*)d_in[1];   // [64,577,576]
    float* out = (float*)d_out;
    float* out_tokens = out;                                     // [64,433,768]
    float* out_R = out + (size_t)BATCH * OUTROWS * DIM;          // [64,433,432]

    // workspace partitioning (all 256B-aligned block sizes)
    char* ws = (char*)d_ws;
    size_t off = 0;
    int*   order = (int*)(ws + off);   off += (size_t)BATCH * PTOK * 4;        // 147456
    int*   t0    = (int*)(ws + off);   off += (size_t)BATCH * SSZ * 4;         // 73728
    int*   t1    = (int*)(ws + off);   off += (size_t)BATCH * SSZ * 4;
    int*   gidx  = (int*)(ws + off);   off += (size_t)BATCH * GSEL * 4;        // 36864
    float* deg   = (float*)(ws + off); off += (size_t)BATCH * SSZ * 4;
    float* W     = (float*)(ws + off); off += (size_t)BATCH * SSZ * SSZ * 4;   // 21.2 MB
    float* skip  = (float*)(ws + off); off += (size_t)BATCH * SSZ * DIM * 4;   // 56.6 MB
    float* sembs = (float*)(ws + off); off += (size_t)BATCH * SSZ * DIM * 4;   // 56.6 MB
    (void)ws_size; (void)in_sizes; (void)n_in; (void)out_size;

    order_kernel<<<BATCH, PTOK, 0, stream>>>(attn, order);
    adj_kernel<<<dim3(SSZ, BATCH), SSZ, 0, stream>>>(attn, order, W);
    gather_tokens_kernel<<<dim3(NTOK, BATCH), 192, 0, stream>>>(tk, order, out_tokens, skip);
    top2_kernel<<<BATCH, SSZ, 0, stream>>>(W, t0, t1);
    mask_topk_kernel<<<dim3(SSZ, BATCH), SSZ, 0, stream>>>(W, t0, t1);
    fixpoint_kernel<<<BATCH, SSZ, 0, stream>>>(W, deg);
    wmma_aggregate_kernel<<<dim3(DIM / 16, SSZ / 16, BATCH), 32, 0, stream>>>(W, skip, sembs);
    group_kernel<<<BATCH, SSZ, 0, stream>>>(deg, gidx);
    gather_sel_kernel<<<dim3(GSEL, BATCH), 192, 0, stream>>>(sembs, gidx, out_tokens);
    build_R_kernel<<<dim3(OUTROWS, BATCH), RCOLS, 0, stream>>>(attn, order, gidx, out_R);
}